// GCN_8650064134273
// MI455X (gfx1250) — compile-verified
//
#include <hip/hip_runtime.h>
#include <hip/hip_bf16.h>

typedef float v2f __attribute__((ext_vector_type(2)));
typedef float v8f __attribute__((ext_vector_type(8)));

#define D_FEAT 128

// ---------------------------------------------------------------------------
// Kernel 1: xtheta[m][n] = sum_k x[m][k] * W[n][k] + b[n]
// One block = 16 rows of x. 8 waves/block, wave w computes 16x16 tile at
// columns [16w, 16w+16). K=128 consumed in 32 steps of V_WMMA_F32_16X16X4_F32.
//
// f32 WMMA fragment layouts (ISA 7.12.2):
//   A (16x4, MxK): lanes 0-15 -> M=lane, VGPR{0,1} = K{0,1};
//                  lanes 16-31 -> M=lane-16, VGPR{0,1} = K{2,3}
//   B (4x16, KxN): mirrored with N across lanes.
//   C (16x16):     lanes 0-15 -> N=lane, VGPR r -> M=r;
//                  lanes 16-31 -> N=lane-16, VGPR r -> M=r+8
// ---------------------------------------------------------------------------
__global__ void __launch_bounds__(256)
gcn_gemm_wmma_f32(const float* __restrict__ x,
                  const float* __restrict__ W,
                  const float* __restrict__ bias,
                  float* __restrict__ xtheta,
                  int nrows)
{
    const int wave  = threadIdx.x >> 5;          // 0..7 -> column tile
    const int lane  = threadIdx.x & 31;
    const int lhalf = lane & 15;                 // M (A) / N (B,C) within tile
    const int khalf = (lane >> 4) << 1;          // 0 or 2: K sub-offset

    const int m0 = blockIdx.x * 16;
    const int n0 = wave * 16;

    int arow = m0 + lhalf;
    if (arow >= nrows) arow = nrows - 1;         // clamp (tail-safe; N%16==0 here)

    const float* __restrict__ aptr = x + (size_t)arow * D_FEAT + khalf;
    const float* __restrict__ bptr = W + (size_t)(n0 + lhalf) * D_FEAT + khalf;

    v8f c = {};
#pragma unroll
    for (int k = 0; k < D_FEAT; k += 4) {
        v2f afrag = *(const v2f*)(aptr + k);     // global_load_b64
        v2f bfrag = *(const v2f*)(bptr + k);
        c = __builtin_amdgcn_wmma_f32_16x16x4_f32(
                /*neg_a=*/false, afrag,
                /*neg_b=*/false, bfrag,
                /*c_mod=*/(short)0, c,
                /*reuse_a=*/false, /*reuse_b=*/false);
    }

    const float bn    = bias[n0 + lhalf];
    const int   mbase = m0 + ((lane >> 4) << 3); // +8 rows for upper lane half
#pragma unroll
    for (int r = 0; r < 8; ++r) {
        const int m = mbase + r;
        if (m < nrows)
            xtheta[(size_t)m * D_FEAT + n0 + lhalf] = c[r] + bn;
    }
}

// ---------------------------------------------------------------------------
// Kernel 2: COO SpMM scatter-add. One wave32 per edge; lane l covers features
// l, l+32, l+64, l+96. Gathers hit L2 (xtheta = 25.6 MB << 192 MB L2);
// accumulation via hardware global_atomic_add_f32 (L2 atomic units).
// ---------------------------------------------------------------------------
__global__ void __launch_bounds__(256)
gcn_spmm_coo(const int*   __restrict__ row,
             const int*   __restrict__ col,
             const float* __restrict__ avals,
             const float* __restrict__ xtheta,
             float*       __restrict__ agg,
             int nedges)
{
    const int lane = threadIdx.x & 31;
    const int e    = blockIdx.x * (blockDim.x >> 5) + (threadIdx.x >> 5);
    if (e >= nedges) return;

    const int   r = row[e];
    const int   s = col[e];
    const float a = avals[e];

    const float* __restrict__ src = xtheta + (size_t)s * D_FEAT;
    float*       __restrict__ dst = agg    + (size_t)r * D_FEAT;

#pragma unroll
    for (int j = 0; j < 4; ++j) {
        const int f = lane + j * 32;
        __hip_atomic_fetch_add(dst + f, a * src[f],
                               __ATOMIC_RELAXED, __HIP_MEMORY_SCOPE_AGENT);
    }
}

// ---------------------------------------------------------------------------
// Kernel 3: PReLU in place on the aggregated output, float4-vectorized.
// ---------------------------------------------------------------------------
__global__ void __launch_bounds__(256)
gcn_prelu_inplace(float* __restrict__ out,
                  const float* __restrict__ alpha,
                  int nvec4)
{
    const int i = blockIdx.x * blockDim.x + threadIdx.x;
    if (i >= nvec4) return;
    const float al = alpha[0];
    float4* p = reinterpret_cast<float4*>(out);
    float4 v = p[i];
    v.x = v.x > 0.0f ? v.x : al * v.x;
    v.y = v.y > 0.0f ? v.y : al * v.y;
    v.z = v.z > 0.0f ? v.z : al * v.z;
    v.w = v.w > 0.0f ? v.w : al * v.w;
    p[i] = v;
}

// ---------------------------------------------------------------------------
// Launcher. Inputs (setup_inputs order): x, row, col, a_vals, W, b, alpha.
// Workspace: xtheta [N*128] floats at d_ws. d_out doubles as the accumulator.
// ---------------------------------------------------------------------------
extern "C" void kernel_launch(void* const* d_in, const int* in_sizes, int n_in,
                              void* d_out, int out_size, void* d_ws, size_t ws_size,
                              hipStream_t stream)
{
    const float* x     = (const float*)d_in[0];
    const int*   row   = (const int*)  d_in[1];
    const int*   col   = (const int*)  d_in[2];
    const float* avals = (const float*)d_in[3];
    const float* W     = (const float*)d_in[4];
    const float* bias  = (const float*)d_in[5];
    const float* alpha = (const float*)d_in[6];
    float* out = (float*)d_out;

    const int N = in_sizes[0] / D_FEAT;   // 50000
    const int E = in_sizes[1];            // 800000

    float* xtheta = (float*)d_ws;         // N*128 floats = 25.6 MB

    // zero the accumulator (d_out) — required every call (atomic accumulate)
    hipMemsetAsync(out, 0, (size_t)N * D_FEAT * sizeof(float), stream);

    // 1) dense projection via fp32 WMMA
    const int row_tiles = (N + 15) / 16;
    gcn_gemm_wmma_f32<<<row_tiles, 256, 0, stream>>>(x, W, bias, xtheta, N);

    // 2) COO scatter-add (8 edges per 256-thread block)
    const int spmm_blocks = (E + 7) / 8;
    gcn_spmm_coo<<<spmm_blocks, 256, 0, stream>>>(row, col, avals, xtheta, out, E);

    // 3) PReLU in place
    const int nvec4 = (N * D_FEAT) / 4;
    gcn_prelu_inplace<<<(nvec4 + 255) / 256, 256, 0, stream>>>(out, alpha, nvec4);
}